// BranchModel_21285857919353
// MI455X (gfx1250) — compile-verified
//
#include <hip/hip_runtime.h>
#include <stdint.h>

#define B_      8
#define T_      524288
#define THREADS 128
#define LSEG    64
#define S_BLK   (THREADS * LSEG)        // 8192 outputs per block
#define WARM    18                      // total FIR lookback
#define HALO    20                      // x halo (>=WARM+2, 16B aligned)
#define XBUF_N  (S_BLK + 32)            // 8224 floats per component
#define NCHUNK  ((S_BLK + HALO) / 4)    // 2053 b128 chunks per component
#define WBUF_N2 (S_BLK + 8)             // w ring: 8 halo + S_BLK, float2
#define XR_OFF  0
#define XI_OFF  XBUF_N
#define W_OFF   (2 * XBUF_N)
#define HR_OFF  (W_OFF + 2 * WBUF_N2)
#define HI_OFF  (HR_OFF + 40)
#define HIN_OFF (HI_OFF + 40)
#define LDS_FLOATS (HIN_OFF + 40)       // 32968 floats = 131,872 B

typedef float v2f __attribute__((ext_vector_type(2)));
typedef float v8f __attribute__((ext_vector_type(8)));
typedef int g_v4i __attribute__((vector_size(16)));   // matches builtin proto

#if defined(__HIP_DEVICE_COMPILE__) && defined(__gfx1250__)
#if __has_builtin(__builtin_amdgcn_global_load_async_to_lds_b128)
#define HAVE_ASY_LD 1
#endif
#if __has_builtin(__builtin_amdgcn_global_store_async_from_lds_b128)
#define HAVE_ASY_ST 1
#endif
#if __has_builtin(__builtin_amdgcn_wmma_f32_16x16x4_f32)
#define HAVE_WMMA 1
#endif
#endif

#define AS1 __attribute__((address_space(1)))
#define AS3 __attribute__((address_space(3)))

#if defined(HAVE_ASY_LD) || defined(HAVE_ASY_ST)
__device__ __forceinline__ void wait_async0() {
#if __has_builtin(__builtin_amdgcn_s_wait_asynccnt)
  __builtin_amdgcn_s_wait_asynccnt(0);
#else
  asm volatile("s_wait_asynccnt 0" ::: "memory");
#endif
}
#endif

__global__ __launch_bounds__(THREADS) void branch_pipeline_kernel(
    const float* __restrict__ xre, const float* __restrict__ xim,
    const float* __restrict__ c1p, const float* __restrict__ f1rp,
    const float* __restrict__ f1ip, const float* __restrict__ c2p,
    const float* __restrict__ f2rp, const float* __restrict__ f2ip,
    const float* __restrict__ houtRe, const float* __restrict__ houtIm,
    float* __restrict__ out) {
  __shared__ float smem[LDS_FLOATS];
  const int tid  = threadIdx.x;
  const int row  = blockIdx.y;
  const int base = blockIdx.x * S_BLK;
  const float* xr = xre + (size_t)row * T_;
  const float* xi = xim + (size_t)row * T_;

  // ---------------- Phase A: stage x (+halo, zero below t=0) into LDS -------
  for (int c = tid; c < NCHUNK; c += THREADS) {
    const int g0 = base - HALO + 4 * c;   // 16B-aligned global float index
    if (g0 >= 0) {
#if defined(HAVE_ASY_LD)
      __builtin_amdgcn_global_load_async_to_lds_b128(
          (AS1 g_v4i*)(AS1 float*)(float*)(xr + g0),
          (AS3 g_v4i*)(AS3 float*)&smem[XR_OFF + 4 * c], 0, 0);
      __builtin_amdgcn_global_load_async_to_lds_b128(
          (AS1 g_v4i*)(AS1 float*)(float*)(xi + g0),
          (AS3 g_v4i*)(AS3 float*)&smem[XI_OFF + 4 * c], 0, 0);
#else
#pragma unroll
      for (int q = 0; q < 4; ++q) {
        smem[XR_OFF + 4 * c + q] = xr[g0 + q];
        smem[XI_OFF + 4 * c + q] = xi[g0 + q];
      }
#endif
    } else {  // entire chunk below t=0 (only for blockIdx.x==0): zero-fill
#pragma unroll
      for (int q = 0; q < 4; ++q) {
        smem[XR_OFF + 4 * c + q] = 0.f;
        smem[XI_OFF + 4 * c + q] = 0.f;
      }
    }
  }
#if defined(HAVE_ASY_LD)
  wait_async0();
#endif
  float2* wbuf = (float2*)&smem[W_OFF];
  if (tid < 8) wbuf[tid] = make_float2(0.f, 0.f);  // front halo slots
  __syncthreads();

  // ---------------- coefficients (uniform) into registers -------------------
  float C1[3][8], F1r[3][5], F1i[3][5];
  float C2[3][5][6], F2r[3][5][3], F2i[3][5][3];
#pragma unroll
  for (int i = 0; i < 3; ++i) {
#pragma unroll
    for (int k = 0; k < 8; ++k) C1[i][k] = c1p[i * 8 + k];
#pragma unroll
    for (int k = 0; k < 5; ++k) {
      F1r[i][k] = f1rp[i * 5 + k];
      F1i[i][k] = f1ip[i * 5 + k];
    }
#pragma unroll
    for (int j = 0; j < 5; ++j) {
#pragma unroll
      for (int k = 0; k < 6; ++k) C2[i][j][k] = c2p[(i * 5 + j) * 6 + k];
#pragma unroll
      for (int k = 0; k < 3; ++k) {
        F2r[i][j][k] = f2rp[(i * 5 + j) * 3 + k];
        F2i[i][j][k] = f2ip[(i * 5 + j) * 3 + k];
      }
    }
  }

  // ---------------- Phase B: sequential scan, scatter-FIR rings -------------
  float XiR[3][5] = {}, XiI[3][5] = {};   // xi[t..t+4] accumulators per i
  float WR[7] = {}, WI[7] = {};           // w[t..t+6] accumulators
  float xm1r, xm1i, xm2r, xm2i;
  {
    const int l1 = tid * LSEG + 1;        // index of x[tstart-1] in LDS
    xm1r = smem[XR_OFF + l1];     xm1i = smem[XI_OFF + l1];
    xm2r = smem[XR_OFF + l1 - 1]; xm2i = smem[XI_OFF + l1 - 1];
  }

#pragma unroll 1
  for (int s = 0; s < LSEG + WARM; ++s) {
    const int lidx = tid * LSEG + 2 + s;  // x[t], t = start-WARM+s
    const float x0r = smem[XR_OFF + lidx];
    const float x0i = smem[XI_OFF + lidx];
    const float ur[3] = {x0r, xm1r, xm2r};
    const float ui[3] = {x0i, xm1i, xm2i};
#pragma unroll
    for (int i = 0; i < 3; ++i) {
      // ---- stage 1: gain + scatter through FIR-5 into xi ring ----
      float rr = __builtin_sqrtf(ur[i] * ur[i] + ui[i] * ui[i]);
      float Tb[8];
      Tb[0] = 1.f; Tb[1] = rr;
      const float r2 = rr + rr;
#pragma unroll
      for (int k = 2; k < 8; ++k) Tb[k] = r2 * Tb[k - 1] - Tb[k - 2];
      float g = 0.f;
#pragma unroll
      for (int k = 0; k < 8; ++k) g += C1[i][k] * Tb[k];
      const float z1r = ur[i] * g, z1i = ui[i] * g;
#pragma unroll
      for (int k = 0; k < 5; ++k) {
        XiR[i][k] += F1r[i][k] * z1r - F1i[i][k] * z1i;
        XiI[i][k] += F1r[i][k] * z1i + F1i[i][k] * z1r;
      }
      // ---- xi[t] complete: stage 2, shared Chebyshev basis over j ----
      const float ar = XiR[i][0], ai = XiI[i][0];
      float r22 = __builtin_sqrtf(ar * ar + ai * ai);
      float Ub[6];
      Ub[0] = 1.f; Ub[1] = r22;
      const float u2 = r22 + r22;
#pragma unroll
      for (int k = 2; k < 6; ++k) Ub[k] = u2 * Ub[k - 1] - Ub[k - 2];
#pragma unroll
      for (int j = 0; j < 5; ++j) {
        float g2 = 0.f;
#pragma unroll
        for (int k = 0; k < 6; ++k) g2 += C2[i][j][k] * Ub[k];
        const float zr = ar * g2, zi = ai * g2;
#pragma unroll
        for (int kp = 0; kp < 3; ++kp) {
          const int d = j + kp;  // z_{i,j}[t+j] scattered through FIR-3
          WR[d] += F2r[i][j][kp] * zr - F2i[i][j][kp] * zi;
          WI[d] += F2r[i][j][kp] * zi + F2i[i][j][kp] * zr;
        }
      }
      // shift xi ring
#pragma unroll
      for (int k = 0; k < 4; ++k) {
        XiR[i][k] = XiR[i][k + 1];
        XiI[i][k] = XiI[i][k + 1];
      }
      XiR[i][4] = 0.f; XiI[i][4] = 0.f;
    }
    xm2r = xm1r; xm2i = xm1i; xm1r = x0r; xm1i = x0i;
    // ---- w[t] complete: store (incl. 6-sample halo for the hout stage) ----
    if (s >= WARM - 6) {
      wbuf[tid * LSEG + s - 10] = make_float2(WR[0], WI[0]);
    }
#pragma unroll
    for (int d = 0; d < 6; ++d) { WR[d] = WR[d + 1]; WI[d] = WI[d + 1]; }
    WR[6] = 0.f; WI[6] = 0.f;
  }
  __syncthreads();

  // ---------------- hout tap tables (zero-padded Toeplitz source) ----------
  if (tid < 40) {
    const int d = tid - 16;
    const float hr = (d >= 0 && d < 7) ? houtRe[d] : 0.f;
    const float hi = (d >= 0 && d < 7) ? houtIm[d] : 0.f;
    smem[HR_OFF + tid] = hr;
    smem[HI_OFF + tid] = hi;
    smem[HIN_OFF + tid] = -hi;  // f32 WMMA has no A-negate modifier
  }
  __syncthreads();

  float2* ybuf = (float2*)&smem[0];  // overlays dead x-stage buffers

#if defined(HAVE_WMMA)
  // ---------------- Phase C: y = FIR(w, hout) via Toeplitz WMMA ------------
  const int lane  = tid & 31;
  const int wv    = tid >> 5;
  const int mrow  = lane & 15;
  const int ncol  = lane & 15;
  const int khalf = (lane >> 4) << 1;  // lanes 16-31 hold K+2 of A / B
  v2f Ahr[6], Ahi[6], Ahin[6];
#pragma unroll
  for (int q = 0; q < 6; ++q) {
    const int tau0 = -8 + 4 * q;
    const int i0 = 16 + mrow - tau0 - khalf;  // in [1,39]
    Ahr[q][0]  = smem[HR_OFF + i0];  Ahr[q][1]  = smem[HR_OFF + i0 - 1];
    Ahi[q][0]  = smem[HI_OFF + i0];  Ahi[q][1]  = smem[HI_OFF + i0 - 1];
    Ahin[q][0] = smem[HIN_OFF + i0]; Ahin[q][1] = smem[HIN_OFF + i0 - 1];
  }
  for (int G = wv; G < S_BLK / 256; G += THREADS / 32) {
    v8f accR = {}, accI = {};
#pragma unroll
    for (int q = 0; q < 6; ++q) {
      const int tau0 = -8 + 4 * q;
      const int bb = 8 + G * 256 + ncol * 16 + tau0 + khalf;  // even -> b128
      const float4 wq = *(const float4*)&wbuf[bb];
      v2f Br, Bi;
      Br[0] = wq.x; Br[1] = wq.z;   // w re at K=khalf, khalf+1
      Bi[0] = wq.y; Bi[1] = wq.w;   // w im
      accR = __builtin_amdgcn_wmma_f32_16x16x4_f32(false, Ahr[q],  false, Br,
                                                   (short)0, accR, false, false);
      accR = __builtin_amdgcn_wmma_f32_16x16x4_f32(false, Ahin[q], false, Bi,
                                                   (short)0, accR, false, false);
      accI = __builtin_amdgcn_wmma_f32_16x16x4_f32(false, Ahr[q],  false, Bi,
                                                   (short)0, accI, false, false);
      accI = __builtin_amdgcn_wmma_f32_16x16x4_f32(false, Ahi[q],  false, Br,
                                                   (short)0, accI, false, false);
    }
    const int tb = G * 256 + ncol * 16 + ((lane >> 4) << 3);
#pragma unroll
    for (int v = 0; v < 8; ++v) ybuf[tb + v] = make_float2(accR[v], accI[v]);
  }
#else
  // ---------------- Phase C fallback: VALU hout FIR ------------------------
  const int start = tid * LSEG;
  for (int t = start; t < start + LSEG; ++t) {
    float yr = 0.f, yq = 0.f;
#pragma unroll
    for (int m = 0; m < 7; ++m) {
      const float hr = houtRe[m], hi = houtIm[m];
      const float2 wv2 = wbuf[t + 8 - m];
      yr += hr * wv2.x - hi * wv2.y;
      yq += hr * wv2.y + hi * wv2.x;
    }
    ybuf[t] = make_float2(yr, yq);
  }
#endif
  __syncthreads();

  // ---------------- Phase D: store interleaved (re,im) output --------------
  float2* outp = (float2*)out + (size_t)row * T_ + base;
#if defined(HAVE_ASY_ST)
  for (int c = tid; c < S_BLK / 2; c += THREADS) {
    __builtin_amdgcn_global_store_async_from_lds_b128(
        (AS1 g_v4i*)(AS1 float2*)(float2*)(outp + 2 * c),
        (AS3 g_v4i*)(AS3 float2*)&ybuf[2 * c], 0, 0);
  }
  wait_async0();
#else
  for (int c = tid; c < S_BLK; c += THREADS) outp[c] = ybuf[c];
#endif
}

extern "C" void kernel_launch(void* const* d_in, const int* in_sizes, int n_in,
                              void* d_out, int out_size, void* d_ws,
                              size_t ws_size, hipStream_t stream) {
  (void)in_sizes; (void)n_in; (void)out_size; (void)d_ws; (void)ws_size;
  dim3 grid(T_ / S_BLK, B_, 1);
  branch_pipeline_kernel<<<grid, THREADS, 0, stream>>>(
      (const float*)d_in[0], (const float*)d_in[1], (const float*)d_in[2],
      (const float*)d_in[3], (const float*)d_in[4], (const float*)d_in[5],
      (const float*)d_in[6], (const float*)d_in[7], (const float*)d_in[8],
      (const float*)d_in[9], (float*)d_out);
}